// EdgeConvEncoder_85873576117019
// MI455X (gfx1250) — compile-verified
//
#include <hip/hip_runtime.h>

#define NC 128
#define NNODES 50000
#define NEDGES 600000
#define NTILES (NNODES / 16)   // 3125, exact

typedef __attribute__((ext_vector_type(2))) float v2f;
typedef __attribute__((ext_vector_type(8))) float v8f;

// ---------------------------------------------------------------------------
// Fold BatchNorm (eval) into affine s,t:  bn(m) = m*s + t
//   s = g * rsqrt(rv + eps),  t = be - rm*s
// ---------------------------------------------------------------------------
__global__ __launch_bounds__(128)
void bnparams_kernel(const float* __restrict__ g1, const float* __restrict__ be1,
                     const float* __restrict__ rm1, const float* __restrict__ rv1,
                     const float* __restrict__ g2, const float* __restrict__ be2,
                     const float* __restrict__ rm2, const float* __restrict__ rv2,
                     float* __restrict__ P /* S1,T1,S2,T2 x128 */) {
    int c = threadIdx.x;
    float s1 = g1[c] * rsqrtf(rv1[c] + 1e-5f);
    float s2 = g2[c] * rsqrtf(rv2[c] + 1e-5f);
    P[c]             = s1;
    P[128 + c]       = be1[c] - rm1[c] * s1;
    P[256 + c]       = s2;
    P[384 + c]       = be2[c] - rm2[c] * s2;
}

// ---------------------------------------------------------------------------
// In-degree of each node (shared by both layers).
// ---------------------------------------------------------------------------
__global__ __launch_bounds__(256)
void degree_kernel(const long long* __restrict__ dst, int* __restrict__ deg) {
    int e = blockIdx.x * 256 + threadIdx.x;
    if (e < NEDGES) atomicAdd(&deg[(int)dst[e]], 1);
}

// ---------------------------------------------------------------------------
// U = X @ (waL - waR)^T ; V = X @ waR^T        (wa: [128, 256] row-major)
// One wave per 16x16 output tile; K=128 via 32x v_wmma_f32_16x16x4_f32.
// Block = 256 threads = 8 waves = 8 column tiles (full 128-wide row of tiles),
// grid = 3125 node tiles (exact, no tail -> EXEC all ones for WMMA).
// A layout (16x4 f32): lanes 0-15 M=0-15 K={0,1}; lanes 16-31 M=0-15 K={2,3}.
// ---------------------------------------------------------------------------
__global__ __launch_bounds__(256)
void gemm_uv_kernel(const float* __restrict__ X, const float* __restrict__ WA,
                    float* __restrict__ U, float* __restrict__ V) {
    const int lane  = threadIdx.x & 31;
    const int wv    = threadIdx.x >> 5;
    const int l15   = lane & 15;
    const int half  = lane >> 4;
    const int m0    = blockIdx.x << 4;
    const int c0    = wv << 4;
    const int khalf = half << 1;

    const float* xr = X  + (size_t)(m0 + l15) * NC;          // A row for this lane
    const float* wr = WA + (size_t)(c0 + l15) * (2 * NC);    // weight row (B^T source)

    v8f accU = {0.f,0.f,0.f,0.f,0.f,0.f,0.f,0.f};
    v8f accV = {0.f,0.f,0.f,0.f,0.f,0.f,0.f,0.f};
#pragma unroll
    for (int k0 = 0; k0 < NC; k0 += 4) {
        const int k = k0 + khalf;
        v2f a  = { xr[k], xr[k + 1] };
        float wl0 = wr[k],      wl1 = wr[k + 1];
        float wr0 = wr[NC + k], wr1 = wr[NC + k + 1];
        v2f bu = { wl0 - wr0, wl1 - wr1 };
        v2f bv = { wr0, wr1 };
        accU = __builtin_amdgcn_wmma_f32_16x16x4_f32(false, a, false, bu, (short)0, accU, false, false);
        accV = __builtin_amdgcn_wmma_f32_16x16x4_f32(false, a, false, bv, (short)0, accV, false, false);
    }
    // D layout: lanes 0-15 -> M = m0+vgpr, lanes 16-31 -> M = m0+8+vgpr; N = c0+l15
    const int mb = m0 + half * 8;
    const int cc = c0 + l15;
#pragma unroll
    for (int r = 0; r < 8; ++r) {
        U[(size_t)(mb + r) * NC + cc] = accU[r];
        V[(size_t)(mb + r) * NC + cc] = accV[r];
    }
}

// ---------------------------------------------------------------------------
// Per-edge: acc[dst] += relu(u[dst] + v[src] + ba) * s + t
// One wave per edge, float4 per lane (128 channels / 32 lanes).
// u/v/acc all fit in L2 (25.6 MB each) -> gathers/atomics are L2-resident.
// ---------------------------------------------------------------------------
__global__ __launch_bounds__(256)
void edge_scatter_kernel(const float* __restrict__ U, const float* __restrict__ V,
                         const long long* __restrict__ src, const long long* __restrict__ dst,
                         const float* __restrict__ BA,
                         const float* __restrict__ S, const float* __restrict__ T,
                         float* __restrict__ ACC) {
    int e = blockIdx.x * 8 + (threadIdx.x >> 5);
    if (e >= NEDGES) return;
    const int lane = threadIdx.x & 31;
    const int i = (int)dst[e];
    const int j = (int)src[e];
    float4 u  = ((const float4*)(U + (size_t)i * NC))[lane];
    float4 v  = ((const float4*)(V + (size_t)j * NC))[lane];
    float4 ba = ((const float4*)BA)[lane];
    float4 s  = ((const float4*)S)[lane];
    float4 t  = ((const float4*)T)[lane];
    float4 m;
    m.x = fmaxf(u.x + v.x + ba.x, 0.f) * s.x + t.x;
    m.y = fmaxf(u.y + v.y + ba.y, 0.f) * s.y + t.y;
    m.z = fmaxf(u.z + v.z + ba.z, 0.f) * s.z + t.z;
    m.w = fmaxf(u.w + v.w + ba.w, 0.f) * s.w + t.w;
    float* a = ACC + (size_t)i * NC + lane * 4;
    unsafeAtomicAdd(a + 0, m.x);     // global_atomic_add_f32
    unsafeAtomicAdd(a + 1, m.y);
    unsafeAtomicAdd(a + 2, m.z);
    unsafeAtomicAdd(a + 3, m.w);
}

// ---------------------------------------------------------------------------
// OUT = ACC @ wb^T + deg*bb  [+ optional relu]   (wb: [128,128] row-major)
// Same WMMA tiling as gemm_uv.
// ---------------------------------------------------------------------------
__global__ __launch_bounds__(256)
void gemm_out_kernel(const float* __restrict__ ACC, const float* __restrict__ WB,
                     const float* __restrict__ BB, const int* __restrict__ DEG,
                     float* __restrict__ OUT, int do_relu) {
    const int lane  = threadIdx.x & 31;
    const int wv    = threadIdx.x >> 5;
    const int l15   = lane & 15;
    const int half  = lane >> 4;
    const int m0    = blockIdx.x << 4;
    const int c0    = wv << 4;
    const int khalf = half << 1;

    const float* ar = ACC + (size_t)(m0 + l15) * NC;
    const float* wr = WB  + (size_t)(c0 + l15) * NC;

    v8f acc = {0.f,0.f,0.f,0.f,0.f,0.f,0.f,0.f};
#pragma unroll
    for (int k0 = 0; k0 < NC; k0 += 4) {
        const int k = k0 + khalf;
        v2f a = { ar[k], ar[k + 1] };
        v2f b = { wr[k], wr[k + 1] };
        acc = __builtin_amdgcn_wmma_f32_16x16x4_f32(false, a, false, b, (short)0, acc, false, false);
    }
    const int mb = m0 + half * 8;
    const int cc = c0 + l15;
    const float bb = BB[cc];
#pragma unroll
    for (int r = 0; r < 8; ++r) {
        float y = acc[r] + (float)DEG[mb + r] * bb;
        if (do_relu) y = fmaxf(y, 0.f);
        OUT[(size_t)(mb + r) * NC + cc] = y;
    }
}

// ---------------------------------------------------------------------------
// Row-wise L2 normalize: h /= max(||h||, 1e-12). One wave per row.
// ---------------------------------------------------------------------------
__global__ __launch_bounds__(256)
void l2norm_kernel(float* __restrict__ H) {
    int row = blockIdx.x * 8 + (threadIdx.x >> 5);
    if (row >= NNODES) return;
    const int lane = threadIdx.x & 31;
    float4* p = (float4*)(H + (size_t)row * NC);
    float4 v = p[lane];
    float ss = v.x * v.x + v.y * v.y + v.z * v.z + v.w * v.w;
#pragma unroll
    for (int off = 16; off > 0; off >>= 1) ss += __shfl_xor(ss, off, 32);
    float inv = 1.0f / fmaxf(sqrtf(ss), 1e-12f);
    v.x *= inv; v.y *= inv; v.z *= inv; v.w *= inv;
    p[lane] = v;
}

// ---------------------------------------------------------------------------
extern "C" void kernel_launch(void* const* d_in, const int* in_sizes, int n_in,
                              void* d_out, int out_size, void* d_ws, size_t ws_size,
                              hipStream_t stream) {
    const float*     x    = (const float*)d_in[0];
    const long long* ei   = (const long long*)d_in[1];   // int64 [2, E]
    const long long* src  = ei;
    const long long* dst  = ei + NEDGES;
    const float* w1a = (const float*)d_in[2];  const float* b1a = (const float*)d_in[3];
    const float* g1  = (const float*)d_in[4];  const float* be1 = (const float*)d_in[5];
    const float* rm1 = (const float*)d_in[6];  const float* rv1 = (const float*)d_in[7];
    const float* w1b = (const float*)d_in[8];  const float* b1b = (const float*)d_in[9];
    const float* w2a = (const float*)d_in[10]; const float* b2a = (const float*)d_in[11];
    const float* g2  = (const float*)d_in[12]; const float* be2 = (const float*)d_in[13];
    const float* rm2 = (const float*)d_in[14]; const float* rv2 = (const float*)d_in[15];
    const float* w2b = (const float*)d_in[16]; const float* b2b = (const float*)d_in[17];
    float* out = (float*)d_out;

    // workspace carve-up (256B aligned slices)
    const size_t NCfl = (size_t)NNODES * NC;
    char* ws = (char*)d_ws;
    size_t off = 0;
    auto take = [&](size_t bytes) { char* p = ws + off; off = (off + bytes + 255) & ~(size_t)255; return p; };
    float* P   = (float*)take(4 * 128 * sizeof(float));  // S1,T1,S2,T2
    int*   deg = (int*)  take(NNODES * sizeof(int));
    float* u   = (float*)take(NCfl * sizeof(float));
    float* v   = (float*)take(NCfl * sizeof(float));
    float* acc = (float*)take(NCfl * sizeof(float));
    float* h   = (float*)take(NCfl * sizeof(float));
    float *S1 = P, *T1 = P + 128, *S2 = P + 256, *T2 = P + 384;
    (void)ws_size; (void)n_in; (void)in_sizes; (void)out_size;

    const int EDGE_BLOCKS = (NEDGES + 7) / 8;

    // prep: BN affine fold + in-degrees
    bnparams_kernel<<<1, 128, 0, stream>>>(g1, be1, rm1, rv1, g2, be2, rm2, rv2, P);
    hipMemsetAsync(deg, 0, NNODES * sizeof(int), stream);
    degree_kernel<<<(NEDGES + 255) / 256, 256, 0, stream>>>(dst, deg);

    // ---- layer 1 ----
    gemm_uv_kernel<<<NTILES, 256, 0, stream>>>(x, w1a, u, v);
    hipMemsetAsync(acc, 0, NCfl * sizeof(float), stream);
    edge_scatter_kernel<<<EDGE_BLOCKS, 256, 0, stream>>>(u, v, src, dst, b1a, S1, T1, acc);
    gemm_out_kernel<<<NTILES, 256, 0, stream>>>(acc, w1b, b1b, deg, h, /*relu=*/1);
    l2norm_kernel<<<(NNODES + 7) / 8, 256, 0, stream>>>(h);

    // ---- layer 2 ----
    gemm_uv_kernel<<<NTILES, 256, 0, stream>>>(h, w2a, u, v);
    hipMemsetAsync(acc, 0, NCfl * sizeof(float), stream);
    edge_scatter_kernel<<<EDGE_BLOCKS, 256, 0, stream>>>(u, v, src, dst, b2a, S2, T2, acc);
    gemm_out_kernel<<<NTILES, 256, 0, stream>>>(acc, w2b, b2b, deg, out, /*relu=*/0);
}